// TopKActivation_22222160790076
// MI455X (gfx1250) — compile-verified
//
#include <hip/hip_runtime.h>
#include <stdint.h>

#define ROWS     4096
#define COLS     24576
#define K        64
#define NTHREADS 256
#define CHUNK    (COLS / NTHREADS)      // 96 contiguous elements per thread
#define VEC4     (COLS / 4 / NTHREADS)  // 24 float4 per thread

// LDS layout in 32-bit words:
//   keys (raw float bits) [0, COLS)
//   hist[256]             [COLS, COLS+256)
//   chunkBase[256]        [COLS+256, COLS+512)
//   selKey[64]            [COLS+512, COLS+576)
//   selCol[64]            [COLS+576, COLS+640)
//   bcast[3]              [COLS+640, COLS+643)  {prefix, remaining, selCount}
#define LDS_WORDS (COLS + 643)

// Map float bits -> unsigned key with same total order as the float values.
__device__ __forceinline__ uint32_t mapkey(uint32_t u) {
    return (u & 0x80000000u) ? ~u : (u | 0x80000000u);
}

#if __has_builtin(__builtin_amdgcn_global_load_async_to_lds_b128) && \
    __has_builtin(__builtin_amdgcn_s_wait_asynccnt)
#define USE_ASYNC_LDS 1
#else
#define USE_ASYNC_LDS 0
#endif

#if USE_ASYNC_LDS
typedef int v4i_t __attribute__((vector_size(16)));
typedef __attribute__((address_space(1))) v4i_t gv4i_t;  // global
typedef __attribute__((address_space(3))) v4i_t lv4i_t;  // LDS
#endif

__global__ __launch_bounds__(NTHREADS)
void topk_scatter_kernel(const float* __restrict__ x,
                         float* __restrict__ out,
                         int* __restrict__ idx_out) {
    extern __shared__ uint32_t lds[];
    uint32_t* keys      = lds;
    uint32_t* hist      = lds + COLS;
    uint32_t* chunkBase = hist + 256;
    uint32_t* selKey    = chunkBase + 256;
    uint32_t* selCol    = selKey + K;
    uint32_t* bcast     = selCol + K;

    const int row = blockIdx.x;
    const int tid = threadIdx.x;
    const float* __restrict__ xrow = x + (size_t)row * COLS;
    float* __restrict__ orow       = out + (size_t)row * COLS;

    // ---- Stage 1: bring the whole row (96 KB) into LDS, raw bits ----
#if USE_ASYNC_LDS
    {
        gv4i_t* gsrc = (gv4i_t*)xrow;   // C-style cast: strips const, adds AS(1)
        lv4i_t* kdst = (lv4i_t*)keys;   // generic->LDS addrspace cast
        for (int i = 0; i < VEC4; ++i) {
            int v = tid + i * NTHREADS;
            __builtin_amdgcn_global_load_async_to_lds_b128(gsrc + v, kdst + v, 0, 0);
        }
        __builtin_amdgcn_s_wait_asynccnt(0);
    }
#else
    for (int i = 0; i < VEC4; ++i) {
        int v = tid + i * NTHREADS;
        float4 f = reinterpret_cast<const float4*>(xrow)[v];
        uint4 u;
        u.x = __float_as_uint(f.x); u.y = __float_as_uint(f.y);
        u.z = __float_as_uint(f.z); u.w = __float_as_uint(f.w);
        reinterpret_cast<uint4*>(keys)[v] = u;
    }
#endif
    if (tid == 0) { bcast[0] = 0u; bcast[1] = K; }
    __syncthreads();

    // ---- Stage 2: 4-pass radix select (MSB first) for k-th largest key ----
    for (int p = 0; p < 4; ++p) {
        hist[tid] = 0u;               // NTHREADS == 256 bins
        __syncthreads();
        const int shift = 24 - 8 * p;
        const uint32_t pref = bcast[0];
        for (int i = tid; i < COLS; i += NTHREADS) {
            uint32_t key = mapkey(keys[i]);
            bool match = (p == 0) ||
                         ((key >> (shift + 8)) == (pref >> (shift + 8)));
            if (match)
                atomicAdd((unsigned int*)&hist[(key >> shift) & 0xFFu], 1u);
        }
        __syncthreads();
        if (tid == 0) {
            uint32_t rem = bcast[1];
            uint32_t cum = 0;
            int b = 0;
            for (int bb = 255; bb >= 0; --bb) {
                uint32_t c = hist[bb];
                if (cum + c >= rem) { b = bb; bcast[1] = rem - cum; break; }
                cum += c;
            }
            bcast[0] = pref | ((uint32_t)b << shift);
        }
        __syncthreads();
    }
    const uint32_t T = bcast[0];   // key of the k-th largest element
    const uint32_t r = bcast[1];   // how many ties at T to keep (lowest index first)

    // ---- Stage 3: index-ordered tie ranking via per-chunk prefix sums ----
    const int base = tid * CHUNK;
    {
        uint32_t myTies = 0;
        for (int j = 0; j < CHUNK; ++j)
            if (mapkey(keys[base + j]) == T) myTies++;
        chunkBase[tid] = myTies;
    }
    __syncthreads();
    if (tid == 0) {
        uint32_t run = 0;
        for (int t = 0; t < NTHREADS; ++t) {
            uint32_t c = chunkBase[t];
            chunkBase[t] = run;
            run += c;
        }
        bcast[2] = 0u;             // selection-list counter
    }
    __syncthreads();

    // ---- Stage 4: scatter result + collect the exactly-64 winners ----
    uint32_t tieRank = chunkBase[tid];
    for (int j = 0; j < CHUNK; ++j) {
        const int col = base + j;
        const uint32_t raw = keys[col];
        const uint32_t key = mapkey(raw);
        bool sel = false;
        if (key > T) {
            sel = true;
        } else if (key == T) {
            if (tieRank < r) sel = true;
            tieRank++;
        }
        orow[col] = sel ? __uint_as_float(raw) : 0.0f;
        if (sel) {
            uint32_t slot = atomicAdd((unsigned int*)&bcast[2], 1u);
            selKey[slot] = key;
            selCol[slot] = (uint32_t)col;
        }
    }
    __syncthreads();

    // ---- Stage 5: rank-sort 64 winners (value desc, index asc) -> idx ----
    if (tid < K) {
        const uint64_t me = ((uint64_t)(~selKey[tid]) << 32) | (uint64_t)selCol[tid];
        int rank = 0;
        for (int j = 0; j < K; ++j) {
            const uint64_t o = ((uint64_t)(~selKey[j]) << 32) | (uint64_t)selCol[j];
            rank += (o < me) ? 1 : 0;
        }
        idx_out[(size_t)row * K + rank] = (int)selCol[tid];
    }
}

extern "C" void kernel_launch(void* const* d_in, const int* in_sizes, int n_in,
                              void* d_out, int out_size, void* d_ws, size_t ws_size,
                              hipStream_t stream) {
    (void)in_sizes; (void)n_in; (void)d_ws; (void)ws_size; (void)out_size;
    const float* x = (const float*)d_in[0];
    // d_in[1] is k (==64), fixed at compile time.
    float* out  = (float*)d_out;
    int* idx    = (int*)((float*)d_out + (size_t)ROWS * COLS);
    const size_t smem = (size_t)LDS_WORDS * sizeof(uint32_t);  // ~101 KB of the 320 KB WGP LDS
    topk_scatter_kernel<<<dim3(ROWS), dim3(NTHREADS), smem, stream>>>(x, out, idx);
}